// Model_2619930051447
// MI455X (gfx1250) — compile-verified
//
#include <hip/hip_runtime.h>

// Swin-style windowed per-head spatial mixing on gfx1250:
//   Y[p, h*8+d] = sum_q Wm[h][p][q] * X[q, h*8+d] + b[h][p]   (4x4 window, C=16, 2 heads)
//
// Bandwidth-bound (512 MB traffic, ~2.15 GFLOP -> ~22us roofline at 23.3 TB/s).
//  * one wave32 <-> one "window pair" = 2048 contiguous bytes of x
//  * input : 4x global_load_async_to_lds_b128  (DMA, ASYNCcnt, no VGPR staging)
//  * B operand built with conflict-free ds gathers from padded LDS stage
//  * 2 heads x 4 chained v_wmma_f32_16x16x4_f32 (f32, K=16), bias preloaded in C
//  * output: D scattered to swizzled LDS stage -> 4x global_store_async_from_lds_b128

typedef float v2f __attribute__((ext_vector_type(2)));
typedef float v8f __attribute__((ext_vector_type(8)));

#define WAVES 8    // 256 threads / wave32
#define PS    20   // padded position-row stride in floats (80B): 16B-aligned rows,
                   // bank-conflict-free B gathers and D scatters (with XOR swizzle)

__global__ __launch_bounds__(256) void swin_head_mix(
    const float* __restrict__ x, const float* __restrict__ Wm,
    const float* __restrict__ bias, float* __restrict__ out, int npairs)
{
  __shared__ __align__(16) float s_in [WAVES][32 * PS]; // [wave][P][c] flat pair layout
  __shared__ __align__(16) float s_out[WAVES][32 * PS]; // same, + XOR-8 channel swizzle

  const int lane = threadIdx.x & 31;
  const int w    = threadIdx.x >> 5;
  const int hi   = lane >> 4;   // K-half select: K += 2*hi (32-bit A/B WMMA layout)
  const int n    = lane & 15;   // B/C/D column index
  const int win  = n >> 3;      // window within the pair (0|1)
  const int hd   = n & 7;       // head-dim column (0..7)

  float* s_inw  = s_in[w];
  float* s_outw = s_out[w];

  // 32-bit LDS byte addresses for the async DMA ops (loop-invariant per lane).
  const unsigned lin  = (unsigned)(uintptr_t)s_inw;
  const unsigned lout = (unsigned)(uintptr_t)s_outw;
  unsigned lds_in[4], lds_out[4];
#pragma unroll
  for (int i = 0; i < 4; ++i) {
    const int j  = i * 32 + lane;         // float4 index in the 2048B pair region
    const int P  = j >> 2;                // position 0..31
    const int cg = j & 3;                 // channel group
    lds_in[i]  = lin + P * (PS * 4) + cg * 16;
    const int cg2 = cg ^ ((P & 16) >> 3); // inverse of the XOR-8 channel swizzle
    lds_out[i] = lout + P * (PS * 4) + cg2 * 16;
  }

  // ---- one-time per-wave: A operands (Wm rows) + bias as initial C ----
  v2f a[2][4];
#pragma unroll
  for (int h = 0; h < 2; ++h)
#pragma unroll
    for (int k = 0; k < 4; ++k) {
      // lane m holds A[m][K], K = 4k + 2*hi + {0,1}
      const float* wp = Wm + (h * 16 + n) * 16 + 4 * k + 2 * hi;
      a[h][k] = *(const v2f*)wp;
    }
  v8f cbias[2];
#pragma unroll
  for (int h = 0; h < 2; ++h) {
    // C/D layout: VGPR v holds row p = v + 8*hi -> bias[h][p]
    const float4 b0 = *(const float4*)(bias + h * 16 + 8 * hi);
    const float4 b1 = *(const float4*)(bias + h * 16 + 8 * hi + 4);
    cbias[h][0] = b0.x; cbias[h][1] = b0.y; cbias[h][2] = b0.z; cbias[h][3] = b0.w;
    cbias[h][4] = b1.x; cbias[h][5] = b1.y; cbias[h][6] = b1.z; cbias[h][7] = b1.w;
  }

  // B-gather base (dwords): element (win,q,c) lives at P(q)*PS + c,
  // P(q) = (q>>2)*8 + win*4 + (q&3); chunk k reads q0 = 4k+2*hi and q0+1.
  const int gbase0 = (4 * win + 2 * hi) * PS + hd;      // + h*8 per head, + 160*k per chunk

  const int stride = (int)gridDim.x * WAVES;
  for (int wpair = (int)blockIdx.x * WAVES + w; wpair < npairs; wpair += stride) {
    const float* src = x   + (size_t)wpair * 512;
    float*       dst = out + (size_t)wpair * 512;

    // ---- async DMA: global -> LDS (4 x 512B). Drain prior ds loads first (WAR on s_in).
    asm volatile("s_wait_dscnt 0x0" ::: "memory");
#pragma unroll
    for (int i = 0; i < 4; ++i) {
      const unsigned long long ga =
          (unsigned long long)(uintptr_t)(src + (i * 32 + lane) * 4);
      asm volatile("global_load_async_to_lds_b128 %0, %1, off"
                   :: "v"(lds_in[i]), "v"(ga) : "memory");
    }
    // Wait DMA complete (also retires last iteration's async stores -> s_out reusable).
    asm volatile("s_wait_asynccnt 0x0" ::: "memory");

    // ---- compute: per head, 4 chained WMMA f32 16x16x4 over K=q ----
    v8f acc[2];
#pragma unroll
    for (int h = 0; h < 2; ++h) {
      acc[h] = cbias[h];
      const int gb = gbase0 + h * 8;
#pragma unroll
      for (int k = 0; k < 4; ++k) {
        const v2f bb = { s_inw[gb + 160 * k], s_inw[gb + 160 * k + PS] };
        acc[h] = __builtin_amdgcn_wmma_f32_16x16x4_f32(
            /*neg_a=*/false, a[h][k], /*neg_b=*/false, bb,
            /*c_mod=*/(short)0, acc[h], /*reuse_a=*/false, /*reuse_b=*/false);
      }
    }

    // ---- scatter D (both heads together so stores pair up), XOR-8 swizzle on P>=16 ----
#pragma unroll
    for (int vv = 0; vv < 8; ++vv) {
      const int p = vv + 8 * hi;
      const int P = ((p >> 2) << 3) | (win << 2) | (p & 3);
      const int swz = (P & 16) >> 1;  // 0 or 8 channels
#pragma unroll
      for (int h = 0; h < 2; ++h)
        s_outw[P * PS + ((h * 8 + hd) ^ swz)] = acc[h][vv];
    }

    // ---- async DMA: LDS -> global (4 x 512B). Scatter must land in LDS first.
    asm volatile("s_wait_dscnt 0x0" ::: "memory");
#pragma unroll
    for (int i = 0; i < 4; ++i) {
      const unsigned long long ga =
          (unsigned long long)(uintptr_t)(dst + (i * 32 + lane) * 4);
      asm volatile("global_store_async_from_lds_b128 %0, %1, off"
                   :: "v"(ga), "v"(lds_out[i]) : "memory");
    }
  }

  // Drain outstanding async stores (S_ENDPGM would also wait-idle, but be explicit).
  asm volatile("s_wait_asynccnt 0x0" ::: "memory");
}

extern "C" void kernel_launch(void* const* d_in, const int* in_sizes, int n_in,
                              void* d_out, int out_size, void* d_ws, size_t ws_size,
                              hipStream_t stream) {
  const float* x  = (const float*)d_in[0];
  const float* Wm = (const float*)d_in[1];
  const float* b  = (const float*)d_in[2];
  float* out      = (float*)d_out;

  const int B      = in_sizes[0] / (64 * 16); // [B, 64, 16] f32
  const int npairs = B * 2;                   // (batch, wy) window pairs

  int blocks = 2048;                          // grid-stride; 8 waves/block
  if (blocks * WAVES > npairs) blocks = (npairs + WAVES - 1) / WAVES;
  if (blocks < 1) blocks = 1;

  swin_head_mix<<<blocks, 256, 0, stream>>>(x, Wm, b, out, npairs);
}